// SparseMOE_62062277427416
// MI455X (gfx1250) — compile-verified
//
#include <hip/hip_runtime.h>
#include <math.h>

#define TT 8192
#define HH 1024
#define EE 8

typedef float v2f __attribute__((ext_vector_type(2)));
typedef float v8f __attribute__((ext_vector_type(8)));

// ---------------------------------------------------------------------------
// Router: logits = x @ gate_w + gate_b ; top-2 + renormalized weights.
// 256 threads = 32 tokens * 8 experts per block.
// ---------------------------------------------------------------------------
__global__ __launch_bounds__(256) void moe_router(
    const float* __restrict__ x, const float* __restrict__ gw,
    const float* __restrict__ gb, float* __restrict__ logits_out,
    int* __restrict__ tok_sel, float* __restrict__ tok_w,
    unsigned* __restrict__ counts)
{
    __shared__ float sm[32][EE];
    const int tid = threadIdx.x;
    const int tl  = tid >> 3;      // local token 0..31
    const int e   = tid & 7;       // expert
    const int token = blockIdx.x * 32 + tl;

    const float* xr = x + (size_t)token * HH;
    float acc = gb[e];
    for (int h = 0; h < HH; h += 4) {
        float4 xv = *(const float4*)(xr + h);
        acc += xv.x * gw[(h + 0) * EE + e];
        acc += xv.y * gw[(h + 1) * EE + e];
        acc += xv.z * gw[(h + 2) * EE + e];
        acc += xv.w * gw[(h + 3) * EE + e];
    }
    logits_out[(size_t)token * EE + e] = acc;
    sm[tl][e] = acc;
    __syncthreads();

    if (tid < 32) {
        const int t2 = blockIdx.x * 32 + tid;
        float l[EE];
#pragma unroll
        for (int i = 0; i < EE; ++i) l[i] = sm[tid][i];
        int i1 = 0;
#pragma unroll
        for (int i = 1; i < EE; ++i) if (l[i] > l[i1]) i1 = i;
        int i2 = (i1 == 0) ? 1 : 0;
#pragma unroll
        for (int i = 0; i < EE; ++i) if (i != i1 && l[i] > l[i2]) i2 = i;
        // renormalized top-2 softmax weights (Z cancels)
        float p2  = expf(l[i2] - l[i1]);
        float inv = 1.0f / (1.0f + p2);
        tok_sel[t2]        = i1 | (i2 << 8);
        tok_w[2 * t2 + 0]  = inv;
        tok_w[2 * t2 + 1]  = p2 * inv;
        atomicAdd(&counts[i1], 1u);
        atomicAdd(&counts[i2], 1u);
    }
}

// ---------------------------------------------------------------------------
// Exclusive prefix over 8 expert counts; zero scatter cursors.
// ---------------------------------------------------------------------------
__global__ void moe_offsets(const unsigned* __restrict__ counts,
                            unsigned* __restrict__ offsets,
                            unsigned* __restrict__ cursor)
{
    if (threadIdx.x == 0 && blockIdx.x == 0) {
        unsigned s = 0;
        for (int e = 0; e < EE; ++e) { offsets[e] = s; s += counts[e]; cursor[e] = 0; }
    }
}

// ---------------------------------------------------------------------------
// Scatter tokens into packed per-expert row lists.
// ---------------------------------------------------------------------------
__global__ __launch_bounds__(256) void moe_scatter(
    const int* __restrict__ tok_sel, const float* __restrict__ tok_w,
    const unsigned* __restrict__ offsets, unsigned* __restrict__ cursor,
    int* __restrict__ rowmap, float* __restrict__ roww)
{
    const int t = blockIdx.x * 256 + threadIdx.x;
    const int sel = tok_sel[t];
#pragma unroll
    for (int j = 0; j < 2; ++j) {
        const int e = (sel >> (8 * j)) & 0xff;
        const unsigned pos = offsets[e] + atomicAdd(&cursor[e], 1u);
        rowmap[pos] = t;
        roww[pos]   = tok_w[2 * t + j];
    }
}

// ---------------------------------------------------------------------------
// FP32 WMMA GEMM over one expert's packed rows.
// Block = 256 threads = 8 waves (wave32). Block tile 32 (M) x 256 (N).
// Wave (mi,ni): mi in {0,1} -> 16 rows, ni in {0..3} -> 64 cols = 4 C frags.
// K staged in LDS in chunks of 32 -> 8 k-steps * 4 wmma per chunk.
//
// B is stored in LDS k-pair-interleaved: element (k, n) at
//   Blds[(k>>1) * BPAD + 2*n + (k&1)]
// so each B fragment (two consecutive K for one column) is a single aligned
// ds_load_b64 into an adjacent VGPR pair -> no register shuffles in the loop.
// BPAD = 544 floats (== 32 mod 64 banks): the upper half-wave reads the next
// k-pair row on the opposite 32 LDS banks -> conflict-free.
// ---------------------------------------------------------------------------
#define BPAD 544

template <bool GATHER, bool RELU, bool SCATTER>
__global__ __launch_bounds__(256) void moe_gemm(
    const float* __restrict__ Asrc,      // GATHER: x[T][H]; else packed h[2T][H]
    const float* __restrict__ W,         // [E][H][H]
    const float* __restrict__ bias,      // [E][H]
    const int* __restrict__ rowmap,
    const float* __restrict__ roww,
    const unsigned* __restrict__ counts,
    const unsigned* __restrict__ offsets,
    float* __restrict__ Dst)             // packed h[2T][H] or (SCATTER) out[T][H]
{
    const int e = blockIdx.z;
    const unsigned cnt  = counts[e];
    const unsigned base = offsets[e];
    const int rt = blockIdx.y;
    if ((unsigned)(rt * 32) >= cnt) return;
    const int nblk = blockIdx.x * 256;
    const float* Wb = W + (size_t)e * HH * HH;

    __shared__ float Alds[32 * 36];       // 32 x 32 tile, pad to 36
    __shared__ float Blds[16 * BPAD];     // 16 k-pair rows x 256 cols x 2 (interleaved)

    const int tid   = threadIdx.x;
    const int lane  = tid & 31;
    const int wave  = tid >> 5;
    const int l15   = lane & 15;
    const int khalf = lane >> 4;          // 0 | 1
    const int koff  = khalf * 2;          // A/B frag K sub-offset
    const int wm    = (wave >> 2) * 16;   // wave M base within block tile
    const int wn    = (wave & 3) * 64;    // wave N base within block tile

    v8f c[4] = {};

    // Per-thread A staging assignment: 4 floats (row r_a, cols c_a..c_a+3)
    const int r_a = tid >> 3;
    const int c_a = (tid & 7) * 4;
    const int rexp = rt * 32 + r_a;
    const bool aval = rexp < (int)cnt;
    const float* arow = nullptr;
    if (aval) {
        if (GATHER) arow = Asrc + (size_t)rowmap[base + rexp] * HH;
        else        arow = Asrc + (size_t)(base + rexp) * HH;
    }
    // B staging: per unit u, thread owns k-pair row rkp, cols f..f+3.
    const int rkp0 = tid >> 6;            // 0..3
    const int bf   = (tid & 63) * 4;      // 0..252

    for (int k0 = 0; k0 < HH; k0 += 32) {
        __syncthreads();
        // ---- stage A (natural layout) ----
        float4 av = make_float4(0.f, 0.f, 0.f, 0.f);
        if (aval) av = *(const float4*)(arow + k0 + c_a);
        *(float4*)(&Alds[r_a * 36 + c_a]) = av;
        // ---- stage B (k-pair interleaved) ----
#pragma unroll
        for (int u = 0; u < 4; ++u) {
            const int rkp = rkp0 + u * 4;            // 0..15
            const int k   = 2 * rkp;
            const float* src = Wb + (size_t)(k0 + k) * HH + nblk + bf;
            float4 r0 = *(const float4*)(src);
            float4 r1 = *(const float4*)(src + HH);
            float* dst = &Blds[rkp * BPAD + 2 * bf];
            ((float4*)dst)[0] = make_float4(r0.x, r1.x, r0.y, r1.y);
            ((float4*)dst)[1] = make_float4(r0.z, r1.z, r0.w, r1.w);
        }
        // prefetch next chunk's weight rows while we compute this one
        if (k0 + 32 < HH) {
            __builtin_prefetch(Wb + (size_t)(k0 + 32 + 2 * rkp0) * HH + nblk + bf, 0, 0);
        }
        __syncthreads();

#pragma unroll
        for (int kk = 0; kk < 32; kk += 4) {
            // A frag: lanes 0-15 hold K=kk,kk+1; lanes 16-31 hold K=kk+2,kk+3
            v2f a = *(const v2f*)(&Alds[(wm + l15) * 36 + kk + koff]);
            const int kp = (kk + koff) >> 1;
#pragma unroll
            for (int j = 0; j < 4; ++j) {
                v2f b = *(const v2f*)(&Blds[kp * BPAD + 2 * (wn + j * 16 + l15)]);
                c[j] = __builtin_amdgcn_wmma_f32_16x16x4_f32(
                    false, a, false, b, (short)0, c[j], false, false);
            }
        }
    }

    // Epilogue: C layout -> lane = column, VGPR i = row (upper half rows 8-15)
    const float* be = bias + (size_t)e * HH;
#pragma unroll
    for (int j = 0; j < 4; ++j) {
        const int n = nblk + wn + j * 16 + l15;
        const float bv = be[n];
#pragma unroll
        for (int i = 0; i < 8; ++i) {
            const int mrow = rt * 32 + wm + khalf * 8 + i;
            if (mrow < (int)cnt) {
                float v = c[j][i] + bv;
                if (RELU) v = fmaxf(v, 0.0f);
                if (SCATTER) {
                    const int tok = rowmap[base + mrow];
                    const float w = roww[base + mrow];
                    atomicAdd(&Dst[(size_t)tok * HH + n], w * v);
                } else {
                    Dst[(size_t)(base + mrow) * HH + n] = v;
                }
            }
        }
    }
}

// ---------------------------------------------------------------------------
extern "C" void kernel_launch(void* const* d_in, const int* in_sizes, int n_in,
                              void* d_out, int out_size, void* d_ws, size_t ws_size,
                              hipStream_t stream)
{
    const float* x  = (const float*)d_in[0];
    const float* gw = (const float*)d_in[1];
    const float* gb = (const float*)d_in[2];
    const float* w1 = (const float*)d_in[3];
    const float* b1 = (const float*)d_in[4];
    const float* w2 = (const float*)d_in[5];
    const float* b2 = (const float*)d_in[6];
    const float* w3 = (const float*)d_in[7];
    const float* b3 = (const float*)d_in[8];

    float* out    = (float*)d_out;                 // [T][H]
    float* logits = out + (size_t)TT * HH;         // [T][E]

    unsigned* counts  = (unsigned*)d_ws;                 // 8
    unsigned* cursor  = counts + 8;                      // 8
    unsigned* offsets = cursor + 8;                      // 8
    int*      tok_sel = (int*)(offsets + 8);             // T
    float*    tok_w   = (float*)(tok_sel + TT);          // 2T
    int*      rowmap  = (int*)(tok_w + 2 * TT);          // 2T
    float*    roww    = (float*)(rowmap + 2 * TT);       // 2T
    float*    h1      = roww + 2 * TT;                   // 2T*H
    float*    h2      = h1 + (size_t)2 * TT * HH;        // 2T*H

    hipMemsetAsync(out, 0, (size_t)TT * HH * sizeof(float), stream);
    hipMemsetAsync(counts, 0, 8 * sizeof(unsigned), stream);

    moe_router<<<TT / 32, 256, 0, stream>>>(x, gw, gb, logits, tok_sel, tok_w, counts);
    moe_offsets<<<1, 1, 0, stream>>>(counts, offsets, cursor);
    moe_scatter<<<TT / 256, 256, 0, stream>>>(tok_sel, tok_w, offsets, cursor, rowmap, roww);

    dim3 grid(HH / 256, TT / 32, EE);   // worst-case rows per expert; blocks early-exit
    moe_gemm<true,  true,  false><<<grid, 256, 0, stream>>>(x,  w1, b1, rowmap, roww, counts, offsets, h1);
    moe_gemm<false, true,  false><<<grid, 256, 0, stream>>>(h1, w2, b2, rowmap, roww, counts, offsets, h2);
    moe_gemm<false, false, true ><<<grid, 256, 0, stream>>>(h2, w3, b3, rowmap, roww, counts, offsets, out);
}